// SPDAvgPooling_2284922602276
// MI455X (gfx1250) — compile-verified
//
#include <hip/hip_runtime.h>

// Weighted Frechet (Karcher) mean of SPD matrices, affine-invariant metric.
// B=256 problems, N=64 members, D=32, maxiter from d_in[2].
// One workgroup (4 waves) per batch element; all state in LDS (64KB);
// all 32x32 fp32 matmuls on V_WMMA_F32_16X16X4_F32; member matrices are
// double-buffered with GLOBAL_LOAD_ASYNC_TO_LDS_B128 so HBM latency hides
// behind the serial Jacobi eigensolves.

#define MD 32            // matrix dim
#define NMAT 64          // members per batch
#define NWAVES 4         // waves per block
#define BLOCKT (NWAVES * 32)
#define NSWEEP 6         // cyclic Jacobi sweeps (fixed -> deterministic)
#define EPSF 1e-6f

typedef __attribute__((ext_vector_type(2))) float v2f;
typedef __attribute__((ext_vector_type(8))) float v8f;

#if defined(__gfx1250__) && __has_builtin(__builtin_amdgcn_global_load_async_to_lds_b128)
#define USE_ASYNC_LDS 1
#else
#define USE_ASYNC_LDS 0
#endif

__device__ __forceinline__ void wait_async0() {
#if defined(__gfx1250__)
#if __has_builtin(__builtin_amdgcn_s_wait_asynccnt)
  __builtin_amdgcn_s_wait_asynccnt(0);
#else
  asm volatile("s_wait_asynccnt 0x0" ::: "memory");
#endif
#endif
}
__device__ __forceinline__ void wait_ds0() {
#if defined(__gfx1250__)
#if __has_builtin(__builtin_amdgcn_s_wait_dscnt)
  __builtin_amdgcn_s_wait_dscnt(0);
#else
  asm volatile("s_wait_dscnt 0x0" ::: "memory");
#endif
#endif
}

// Async copy of one 32x32 fp32 matrix (4KB) global -> LDS.
// 8 x b128 per lane; tracked on ASYNCcnt (wait with wait_async0()).
#if USE_ASYNC_LDS
typedef int v4i_ __attribute__((vector_size(16)));  // matches builtin prototype
__device__ __forceinline__ void async_load_mat(const float* gsrc, float* ldst, int lane) {
  v4i_* g = (v4i_*)gsrc + lane;   // C-style cast drops const per prototype
  v4i_* l = (v4i_*)ldst + lane;
#pragma unroll
  for (int i = 0; i < 8; ++i)
    __builtin_amdgcn_global_load_async_to_lds_b128(g + i * 32, l + i * 32, 0, 0);
}
#else
__device__ __forceinline__ void async_load_mat(const float* gsrc, float* ldst, int lane) {
  const float4* p4 = (const float4*)gsrc;
  float4* a4 = (float4*)ldst;
#pragma unroll
  for (int i = 0; i < 8; ++i) a4[lane + 32 * i] = p4[lane + 32 * i];
}
#endif

// ---- WMMA fragment loaders (32-bit A 16x4 / B 4x16 layouts, wave32) ----
__device__ __forceinline__ v2f ldA(const float* A, int row0, int k0, int m, int h) {
  v2f a;
  a.x = A[(row0 + m) * MD + k0 + 2 * h];
  a.y = A[(row0 + m) * MD + k0 + 2 * h + 1];
  return a;
}
__device__ __forceinline__ v2f ldB(const float* B, int k0, int col0, int m, int h) {
  v2f b;
  b.x = B[(k0 + 2 * h) * MD + col0 + m];
  b.y = B[(k0 + 2 * h + 1) * MD + col0 + m];
  return b;
}
// B accessed transposed: Bt[k][n] = B[n][k]
__device__ __forceinline__ v2f ldBT(const float* B, int k0, int col0, int m, int h) {
  v2f b;
  b.x = B[(col0 + m) * MD + k0 + 2 * h];
  b.y = B[(col0 + m) * MD + k0 + 2 * h + 1];
  return b;
}

// C(32x32) = A(32x32) * B(32x32)   (or A * B^T if TB), one full wave.
template <bool TB>
__device__ __forceinline__ void wave_mm32(const float* A, const float* B, float* C) {
  const int lane = threadIdx.x & 31;
  const int m = lane & 15, h = lane >> 4;
#pragma unroll
  for (int ti = 0; ti < 2; ++ti)
#pragma unroll
    for (int tj = 0; tj < 2; ++tj) {
      v8f c = {};
#pragma unroll
      for (int k0 = 0; k0 < MD; k0 += 4) {
        v2f a = ldA(A, 16 * ti, k0, m, h);
        v2f b = TB ? ldBT(B, k0, 16 * tj, m, h) : ldB(B, k0, 16 * tj, m, h);
        c = __builtin_amdgcn_wmma_f32_16x16x4_f32(false, a, false, b,
                                                  (short)0, c, false, false);
      }
#pragma unroll
      for (int r = 0; r < 8; ++r)
        C[(16 * ti + 8 * h + r) * MD + 16 * tj + m] = c[r];
    }
}

// T += w * (A * B^T), accumulating straight from WMMA fragments via ds_add_f32.
__device__ __forceinline__ void wave_mm32_nt_atomic(const float* A, const float* B,
                                                    float* T, float w) {
  const int lane = threadIdx.x & 31;
  const int m = lane & 15, h = lane >> 4;
#pragma unroll
  for (int ti = 0; ti < 2; ++ti)
#pragma unroll
    for (int tj = 0; tj < 2; ++tj) {
      v8f c = {};
#pragma unroll
      for (int k0 = 0; k0 < MD; k0 += 4) {
        v2f a = ldA(A, 16 * ti, k0, m, h);
        v2f b = ldBT(B, k0, 16 * tj, m, h);
        c = __builtin_amdgcn_wmma_f32_16x16x4_f32(false, a, false, b,
                                                  (short)0, c, false, false);
      }
#pragma unroll
      for (int r = 0; r < 8; ++r)
        atomicAdd(&T[(16 * ti + 8 * h + r) * MD + 16 * tj + m], w * c[r]);
    }
}

// Cyclic Jacobi eigensolver, one wave, matrix M and eigenvectors V in LDS.
// Rotation scalars are broadcast LDS reads -> wave-uniform control flow.
__device__ __forceinline__ void wave_eigh32(float* M, float* V) {
  const int lane = threadIdx.x & 31;
  for (int i = lane; i < MD * MD; i += 32) V[i] = 0.0f;
  V[lane * MD + lane] = 1.0f;
  for (int sweep = 0; sweep < NSWEEP; ++sweep) {
    for (int p = 0; p < MD - 1; ++p) {
      for (int q = p + 1; q < MD; ++q) {
        float apq = M[p * MD + q];             // uniform across lanes
        if (fabsf(apq) > 1e-12f) {
          float app = M[p * MD + p];
          float aqq = M[q * MD + q];
          float theta = (aqq - app) / (2.0f * apq);
          float t = 1.0f / (fabsf(theta) + sqrtf(theta * theta + 1.0f));
          t = (theta >= 0.0f) ? t : -t;
          float cc = rsqrtf(t * t + 1.0f);
          float ss = t * cc;
          // rows p,q (lane handles its column)
          float mp = M[p * MD + lane], mq = M[q * MD + lane];
          M[p * MD + lane] = cc * mp - ss * mq;
          M[q * MD + lane] = ss * mp + cc * mq;
          // cols p,q (lane handles its row)
          float np = M[lane * MD + p], nq = M[lane * MD + q];
          M[lane * MD + p] = cc * np - ss * nq;
          M[lane * MD + q] = ss * np + cc * nq;
          // eigenvector columns p,q
          float vp = V[lane * MD + p], vq = V[lane * MD + q];
          V[lane * MD + p] = cc * vp - ss * vq;
          V[lane * MD + q] = ss * vp + cc * vq;
        }
      }
    }
  }
}

__device__ __forceinline__ void wave_copy(const float* S, float* Dst) {
  const int lane = threadIdx.x & 31;
  for (int i = lane; i < MD * MD; i += 32) Dst[i] = S[i];
}
// O[i][j] = V[i][j] * fe  where fe is this lane's (column j == lane) scale.
__device__ __forceinline__ void wave_scale_cols(const float* V, float fe, float* O) {
  const int lane = threadIdx.x & 31;
  for (int idx = lane; idx < MD * MD; idx += 32) O[idx] = V[idx] * fe;
}
__device__ __forceinline__ void wave_symmetrize(float* A) {
  const int lane = threadIdx.x & 31;
  for (int idx = lane; idx < MD * MD; idx += 32) {
    int i = idx >> 5, j = idx & 31;
    if (i < j) {
      float s = 0.5f * (A[idx] + A[j * MD + i]);
      A[idx] = s;
      A[j * MD + i] = s;
    }
  }
}

__global__ __launch_bounds__(BLOCKT) void frechet_spd_kernel(
    const float* __restrict__ P,   // [B, 64, 32, 32]
    const float* __restrict__ w,   // [64]
    const int* __restrict__ mi,    // [1]  maxiter
    float* __restrict__ out,       // [B, 32, 32]
    int nB) {
  __shared__ float sX[MD * MD];
  __shared__ float sSq[MD * MD];
  __shared__ float sIsq[MD * MD];
  __shared__ float sT[MD * MD];
  __shared__ float sScr[NWAVES][3][MD * MD];   // per-wave {P, V, A}

  const int b = blockIdx.x;
  if (b >= nB) return;
  const int tid = threadIdx.x;
  const int wid = tid >> 5;
  const int lane = tid & 31;
  const size_t base = (size_t)b * NMAT * MD * MD;
  const int maxiter = mi[0];

  // normalized convex weights: every thread computes 1/sum(w) (uniform scalar loads)
  float wsum = 0.0f;
  for (int n = 0; n < NMAT; ++n) wsum += w[n];
  const float inv_wsum = 1.0f / wsum;

  // X0 = sum_n w_n P_n (Euclidean weighted mean)
  for (int idx = tid; idx < MD * MD; idx += BLOCKT) {
    float acc = 0.0f;
    for (int n = 0; n < NMAT; ++n)
      acc += (w[n] * inv_wsum) * P[base + (size_t)n * MD * MD + idx];
    sX[idx] = acc;
  }
  __syncthreads();

  float* PB = sScr[wid][0];
  float* VB = sScr[wid][1];
  float* AB = sScr[wid][2];

  for (int it = 0; it < maxiter; ++it) {
    // zero tangent accumulator (runs concurrently with wave0's eigh below)
    for (int idx = tid; idx < MD * MD; idx += BLOCKT) sT[idx] = 0.0f;

    // kick off this wave's first member load; hides behind the X-eigh
    async_load_mat(P + base + (size_t)wid * MD * MD, PB, lane);

    // wave 0: eigh(X) -> X^{1/2}, X^{-1/2} (uses its V/A buffers, not PB)
    if (wid == 0) {
      float* A0 = sScr[0][2];
      float* V0 = sScr[0][1];
      wave_copy(sX, A0);
      wave_eigh32(A0, V0);
      float e0 = fmaxf(A0[lane * MD + lane], EPSF);
      float fs = sqrtf(e0), fr = rsqrtf(e0);
      wave_scale_cols(V0, fs, A0);       // V diag(sqrt e)
      wave_mm32<true>(A0, V0, sSq);      // * V^T
      wave_scale_cols(V0, fr, A0);       // V diag(1/sqrt e)
      wave_mm32<true>(A0, V0, sIsq);     // * V^T
    }
    __syncthreads();

    // all waves: whiten member, matrix log, accumulate tangent mean.
    // P is double-buffered: next member streams into PB via async-to-LDS
    // while the current member's Jacobi eigensolve runs.
    for (int n = wid; n < NMAT; n += NWAVES) {
      wait_async0();                      // PB holds P_n
      wave_mm32<false>(sIsq, PB, VB);     // VB = X^{-1/2} P
      wait_ds0();                         // all reads of PB retired
      if (n + NWAVES < NMAT)              // stream next member into PB
        async_load_mat(P + base + (size_t)(n + NWAVES) * MD * MD, PB, lane);
      wave_mm32<false>(VB, sIsq, AB);     // AB = whitened M
      wave_symmetrize(AB);
      wave_eigh32(AB, VB);                // long serial section; load in flight
      float e = fmaxf(AB[lane * MD + lane], EPSF);
      float fl = __logf(e);
      wave_scale_cols(VB, fl, AB);        // AB = V diag(log e)
      float wn = w[n] * inv_wsum;         // uniform scalar load
      wave_mm32_nt_atomic(AB, VB, sT, wn); // T += w_n * AB * VB^T
    }
    __syncthreads();

    // wave 0: X <- X^{1/2} expm(T) X^{1/2}
    if (wid == 0) {
      float* A0 = sScr[0][2];
      float* V0 = sScr[0][1];
      wave_copy(sT, A0);
      wave_eigh32(A0, V0);
      float e0 = A0[lane * MD + lane];
      float fe = __expf(e0);
      wave_scale_cols(V0, fe, A0);       // V diag(exp e)
      wave_mm32<true>(A0, V0, sT);       // sT = expm(T)
      wave_mm32<false>(sSq, sT, A0);     // A0 = X^{1/2} expm(T)
      wave_mm32<false>(A0, sSq, sX);     // X  = A0 X^{1/2}
      wave_symmetrize(sX);
    }
    __syncthreads();
  }

  // write result
  for (int idx = tid; idx < MD * MD; idx += BLOCKT)
    out[(size_t)b * MD * MD + idx] = sX[idx];
}

extern "C" void kernel_launch(void* const* d_in, const int* in_sizes, int n_in,
                              void* d_out, int out_size, void* d_ws, size_t ws_size,
                              hipStream_t stream) {
  (void)n_in; (void)d_ws; (void)ws_size; (void)out_size;
  const float* P = (const float*)d_in[0];   // [B,64,32,32] fp32
  const float* w = (const float*)d_in[1];   // [64] fp32
  const int* mi = (const int*)d_in[2];      // [1] maxiter
  float* out = (float*)d_out;               // [B,32,32] fp32
  const int nmat = in_sizes[1];             // 64
  const int nB = in_sizes[0] / (nmat * MD * MD);
  frechet_spd_kernel<<<dim3(nB), dim3(BLOCKT), 0, stream>>>(P, w, mi, out, nB);
}